// Attention_45148696216373
// MI455X (gfx1250) — compile-verified
//
#include <hip/hip_runtime.h>

// Problem constants (from reference): B=16, S=8192, DIM=128, H=4, KV=2, D=64
// Tokens = 131072. Output = [B, 2S, 128] f32 (row = 2*token + t).

typedef float v2f __attribute__((ext_vector_type(2)));
typedef float v8f __attribute__((ext_vector_type(8)));

#define TOK_PER_WG 64
#define NTILES     2048  // 131072 / 64
#define GRID_WGS   1024  // persistent: 2 tiles per WG, double-buffered X
#define XSTR   132       // X tile row stride (floats): bank spread, 16B aligned
#define QKVSTR 65        // feature-major QKV: [feat][tok], odd stride -> bank-free
#define O2STR  132       // out2 row stride

__global__ __launch_bounds__(256, 1)
void fused_attn_f32(const float* __restrict__ x,
                    const float* __restrict__ Wq, const float* __restrict__ bq,
                    const float* __restrict__ Wk, const float* __restrict__ bk,
                    const float* __restrict__ Wv, const float* __restrict__ bv,
                    const float* __restrict__ Wo,
                    float* __restrict__ out)
{
    __shared__ float lds_x[2][TOK_PER_WG * XSTR];     // 2 x 8448 f (double buffer)
    __shared__ float lds_qkv[512 * QKVSTR];           // 33280 f (Q:0..255 K:256..383 V:384..511)
    __shared__ float lds_attn[8 * 64];                //   512 f  [hg][tok]
    __shared__ float lds_out2[128 * O2STR];           // 16896 f

    const int thr  = threadIdx.x;
    const int lane = thr & 31;
    const int wave = thr >> 5;     // 0..7
    const int l16  = lane & 15;
    const int hi   = lane >> 4;    // 0: lanes 0-15, 1: lanes 16-31
    const int klo  = hi * 2;       // K offset inside group-of-4 (A/B frag layout)

    // Async-stage one X tile [64][128] into lds_x[buf] (8 x b128 per thread).
    // LDS dest VGPR = low 32 bits of the generic shared pointer.
    auto issue_x_load = [&](int tile, int buf) {
#pragma unroll
        for (int it = 0; it < 8; ++it) {
            int f4 = thr + it * 256;            // 0..2047 float4 slots
            int t  = f4 >> 5;                   // token in tile
            int k4 = (f4 & 31) << 2;            // feature 0..124 step 4
            unsigned lo = (unsigned)(unsigned long long)
                          (const void*)&lds_x[buf][t * XSTR + k4];
            const float* ga = x + ((long)tile * TOK_PER_WG + t) * 128 + k4;
            asm volatile("global_load_async_to_lds_b128 %0, %1, off"
                         :: "v"(lo), "v"(ga) : "memory");
        }
    };

    int cur = 0;
    issue_x_load(blockIdx.x, 0);                // prologue: first tile in flight

    for (int tile = blockIdx.x; tile < NTILES; tile += gridDim.x) {
        // Wait own async loads, then rendezvous so all waves' data is visible.
        asm volatile("s_wait_asynccnt 0x0" ::: "memory");
        __syncthreads();

        // Prefetch next tile's X into the other buffer; overlaps all compute below.
        const int nxt = tile + gridDim.x;
        if (nxt < NTILES) issue_x_load(nxt, cur ^ 1);

        const float* xt  = &lds_x[cur][0];
        const long  tok0 = (long)tile * TOK_PER_WG;

        // ------------- Phase 1: QKV = X @ WE^T + bE  (WMMA f32 16x16x4) -------
        // 32 N-columns of 16 features: col 0..15 -> Q, 16..23 -> K, 24..31 -> V.
        for (int ci = 0; ci < 4; ++ci) {
            const int col = wave + ci * 8;          // wave-uniform
            const float* wrow;                      // remapped weight row for lane's N
            float bval;
            int f0;
            if (col < 16) {                         // Q : WqE[n] = Wq[n%4][n/4]
                int n = col * 16 + l16;
                wrow = Wq + (n & 3) * (64 * 128) + (n >> 2) * 128;
                bval = bq[(n & 3) * 64 + (n >> 2)];
                f0   = col * 16;
            } else if (col < 24) {                  // K : WkE[n] = Wk[n%2][n/2]
                int n = (col - 16) * 16 + l16;
                wrow = Wk + (n & 1) * (64 * 128) + (n >> 1) * 128;
                bval = bk[(n & 1) * 64 + (n >> 1)];
                f0   = 256 + (col - 16) * 16;
            } else {                                // V
                int n = (col - 24) * 16 + l16;
                wrow = Wv + (n & 1) * (64 * 128) + (n >> 1) * 128;
                bval = bv[(n & 1) * 64 + (n >> 1)];
                f0   = 384 + (col - 24) * 16;
            }

            v2f b[32];                              // B frags for whole K loop (64 VGPRs)
#pragma unroll
            for (int kk = 0; kk < 32; ++kk)
                b[kk] = *(const v2f*)(wrow + kk * 4 + klo);

            for (int ms = 0; ms < 4; ++ms) {
                v8f acc;
#pragma unroll
                for (int r = 0; r < 8; ++r) acc[r] = bval;   // bias pre-load

                const float* ax = xt + (ms * 16 + l16) * XSTR + klo;
#pragma unroll
                for (int kk = 0; kk < 32; ++kk) {
                    v2f a = *(const v2f*)(ax + kk * 4);       // ds_load_b64
                    acc = __builtin_amdgcn_wmma_f32_16x16x4_f32(
                              false, a, false, b[kk], (short)0, acc, false, false);
                }
                const int tokb = ms * 16 + hi * 8;   // D: VGPR r -> M = r + 8*hi
                const int feat = f0 + l16;
#pragma unroll
                for (int r = 0; r < 8; ++r)
                    lds_qkv[feat * QKVSTR + tokb + r] = acc[r];
            }
        }
        __syncthreads();

        // ------------- Phase 2a: qmean over heads, scale K in place -----------
#pragma unroll 2
        for (int it = 0; it < 16; ++it) {
            int idx = thr + it * 256;               // (d, tok): 64*64
            int t = idx & 63;
            int d = idx >> 6;
            float q0 = lds_qkv[(0 * 64 + d) * QKVSTR + t];
            float q1 = lds_qkv[(1 * 64 + d) * QKVSTR + t];
            float q2 = lds_qkv[(2 * 64 + d) * QKVSTR + t];
            float q3 = lds_qkv[(3 * 64 + d) * QKVSTR + t];
            float qm = 0.25f * (q0 + q1 + q2 + q3);
            lds_qkv[(256 +      d) * QKVSTR + t] *= qm;
            lds_qkv[(256 + 64 + d) * QKVSTR + t] *= qm;
        }
        __syncthreads();

        // ------------- Phase 2b: attn[h][g] = Q[h,:].Ks[g,:] / 8 --------------
#pragma unroll
        for (int it = 0; it < 2; ++it) {
            int idx = thr + it * 256;               // 512 = 64 tok * 8 (h,g)
            int t  = idx & 63;
            int hg = idx >> 6;
            int h = hg >> 1, g = hg & 1;
            float acc = 0.f;
#pragma unroll 8
            for (int d = 0; d < 64; ++d)
                acc += lds_qkv[(h * 64 + d) * QKVSTR + t] *
                       lds_qkv[(256 + g * 64 + d) * QKVSTR + t];
            lds_attn[hg * 64 + t] = acc * 0.125f;   // 1/sqrt(64)
        }
        __syncthreads();

        // ------------- Phase 2c: out2 rows = attn @ V, split 256 -> 2x128 -----
        for (int t = 0; t < 64; ++t) {
            int j = thr;                            // j = h*64 + d
            int h = j >> 6, d = j & 63;
            float a0 = lds_attn[(h * 2 + 0) * 64 + t];
            float a1 = lds_attn[(h * 2 + 1) * 64 + t];
            float v0 = lds_qkv[(384 +      d) * QKVSTR + t];
            float v1 = lds_qkv[(384 + 64 + d) * QKVSTR + t];
            lds_out2[(t * 2 + (j >> 7)) * O2STR + (j & 127)] = a0 * v0 + a1 * v1;
        }
        __syncthreads();

        // ------------- Phase 3: y = out2 @ Wo^T (WMMA), store to global -------
        {
            const int nt = wave;                    // one N-tile per wave
            const float* wrow = Wo + (nt * 16 + l16) * 128;
            v2f b[32];
#pragma unroll
            for (int kk = 0; kk < 32; ++kk)
                b[kk] = *(const v2f*)(wrow + kk * 4 + klo);

            for (int mt = 0; mt < 8; ++mt) {
                v8f acc = {};
                const float* ar = &lds_out2[(mt * 16 + l16) * O2STR + klo];
#pragma unroll
                for (int kk = 0; kk < 32; ++kk) {
                    v2f a = *(const v2f*)(ar + kk * 4);
                    acc = __builtin_amdgcn_wmma_f32_16x16x4_f32(
                              false, a, false, b[kk], (short)0, acc, false, false);
                }
                const long rowg = tok0 * 2 + mt * 16 + hi * 8;
                float* op = out + rowg * 128 + nt * 16 + l16;
#pragma unroll
                for (int r = 0; r < 8; ++r)
                    op[(long)r * 128] = acc[r];     // 64B-contiguous per 16-lane half
            }
        }

        cur ^= 1;
    }
}

extern "C" void kernel_launch(void* const* d_in, const int* in_sizes, int n_in,
                              void* d_out, int out_size, void* d_ws, size_t ws_size,
                              hipStream_t stream)
{
    (void)in_sizes; (void)n_in; (void)d_ws; (void)ws_size; (void)out_size;
    const float* x  = (const float*)d_in[0];
    const float* Wq = (const float*)d_in[1];
    const float* bq = (const float*)d_in[2];
    const float* Wk = (const float*)d_in[3];
    const float* bk = (const float*)d_in[4];
    const float* Wv = (const float*)d_in[5];
    const float* bv = (const float*)d_in[6];
    const float* Wo = (const float*)d_in[7];
    float* y = (float*)d_out;

    dim3 grid(GRID_WGS), block(256);
    fused_attn_f32<<<grid, block, 0, stream>>>(x, Wq, bq, Wk, bk, Wv, bv, Wo, y);
}